// WaveletTransform_8237747273759
// MI455X (gfx1250) — compile-verified
//
#include <hip/hip_runtime.h>
#include <hip/hip_bf16.h>

// ---------------------------------------------------------------------------
// 5-level db4 analysis DWT, 64 rows x 262144 fp32.
// Memory-bound (AI ~2 FLOP/B vs 23.3 TB/s HBM; working set fits 192MB L2).
// CDNA5 path used: async global->LDS gather (ASYNCcnt) + LDS tap reuse (16x).
// ---------------------------------------------------------------------------

#define DWT_NT   256                  // threads per block (8 x wave32)
#define DWT_TO   1024                 // output positions per block
#define DWT_SPAN (2 * DWT_TO + 8)     // padded input window per block (floats)

__device__ __forceinline__ void async_load_f32_to_lds(unsigned lds_byte_off,
                                                      const float* gsrc) {
  // GLOBAL_LOAD_ASYNC_TO_LDS_B32: copies one dword from gsrc into LDS at lds_byte_off
  asm volatile("global_load_async_to_lds_b32 %0, %1, off"
               :: "v"(lds_byte_off), "v"(gsrc)
               : "memory");
}

__device__ __forceinline__ void wait_async0() {
  asm volatile("s_wait_asynccnt 0" ::: "memory");
}

__global__ __launch_bounds__(DWT_NT) void dwt_level_kernel(
    const float* __restrict__ x,      // input rows, 64 x L
    const float* __restrict__ fLo,    // 8-tap low-pass
    const float* __restrict__ fHi,    // 8-tap high-pass
    float* __restrict__ loOut,        // 64 x O approximation (next level input)
    float* __restrict__ hiOut,        // 64 x O detail (final output slice)
    int L, int O)
{
  __shared__ float sx[DWT_SPAN];
  __shared__ float sk[16];

  const int tid = threadIdx.x;
  const int b   = blockIdx.y;
  const int j0  = blockIdx.x * DWT_TO;
  const float* xb = x + (size_t)b * (size_t)L;

  if (tid < 16) sk[tid] = (tid < 8) ? fLo[tid] : fHi[tid - 8];

  // Stage padded window p = [2*j0, 2*j0 + SPAN) into LDS via async gather.
  // Padded index p maps to source index q = p - 4 with np-'reflect' folding:
  // q<0 -> -q; q>=L -> 2L-2-q. Only the first/last block of a row reflects.
  const unsigned sxBase = (unsigned)(size_t)(const void*)(&sx[0]);
  const int q0 = 2 * j0 - 4;                       // source index of LDS slot 0
  const bool interior = (q0 >= 0) && (q0 + DWT_SPAN <= L);

  if (interior) {
    // Fast path: identity map, contiguous gather, no reflect math.
    const float* g0 = xb + q0 + tid;
#pragma unroll
    for (int s0 = 0; s0 < DWT_SPAN; s0 += DWT_NT) {
      int s = s0 + tid;
      if (s >= DWT_SPAN) { s = DWT_SPAN - 1; }     // dup-load tail, keep lanes on
      async_load_f32_to_lds(sxBase + 4u * (unsigned)s,
                            (s0 + tid < DWT_SPAN) ? (g0 + s0) : (xb + q0 + s));
    }
  } else {
    // Boundary path: per-lane reflect-mapped gather.
    for (int s0 = 0; s0 < DWT_SPAN; s0 += DWT_NT) {
      int s = s0 + tid;
      if (s >= DWT_SPAN) s = DWT_SPAN - 1;
      int q = q0 + s;
      if (q < 0)  q = -q;
      if (q >= L) q = 2 * L - 2 - q;
      q = (q < 0) ? 0 : ((q >= L) ? (L - 1) : q);  // safety clamp (tail blocks)
      async_load_f32_to_lds(sxBase + 4u * (unsigned)s, xb + q);
    }
  }
  wait_async0();        // per-wave ASYNCcnt drain
  __syncthreads();      // make all waves' LDS writes visible

  float klo[8], khi[8];
#pragma unroll
  for (int k = 0; k < 8; ++k) { klo[k] = sk[k]; khi[k] = sk[8 + k]; }

  const size_t rowOff = (size_t)b * (size_t)O;
#pragma unroll
  for (int it = 0; it < DWT_TO; it += DWT_NT) {
    int j = j0 + it + tid;
    if (j < O) {
      int s = 2 * (it + tid);          // even -> 8B-aligned pairs, conflict-free
      float accLo = 0.f, accHi = 0.f;
#pragma unroll
      for (int k = 0; k < 8; ++k) {
        float v = sx[s + k];
        accLo = fmaf(klo[k], v, accLo);
        accHi = fmaf(khi[k], v, accHi);
      }
      loOut[rowOff + j] = accLo;
      hiOut[rowOff + j] = accHi;
    }
  }
}

extern "C" void kernel_launch(void* const* d_in, const int* in_sizes, int n_in,
                              void* d_out, int out_size, void* d_ws, size_t ws_size,
                              hipStream_t stream) {
  (void)n_in; (void)out_size; (void)ws_size;
  const float* x   = (const float*)d_in[0];
  const float* fLo = (const float*)d_in[1];
  const float* fHi = (const float*)d_in[2];
  float* out = (float*)d_out;
  float* ws  = (float*)d_ws;

  const int B = 64;
  int L = in_sizes[0] / B;             // 262144

  // Ping-pong low-pass buffers in workspace.
  // bufA worst case: B*(L/2+1) floats (level-1 approx).
  size_t aElems = (size_t)B * (size_t)(L / 2 + 1);
  aElems = (aElems + 63) & ~(size_t)63;          // keep bufB 256B-aligned
  float* bufA = ws;
  float* bufB = ws + aElems;

  const float* cur = x;
  float* loDst = bufA;
  size_t outOff = 0;

  for (int lvl = 0; lvl < 5; ++lvl) {
    int O = L / 2 + 1;                 // floor((L+8-8)/2)+1
    dim3 grid((unsigned)((O + DWT_TO - 1) / DWT_TO), (unsigned)B);
    dwt_level_kernel<<<grid, DWT_NT, 0, stream>>>(
        cur, fLo, fHi, loDst, out + outOff, L, O);
    outOff += (size_t)B * (size_t)O;
    cur   = loDst;
    loDst = (lvl % 2 == 0) ? bufB : bufA;
    L = O;
  }
}